// GAT_83700322664814
// MI455X (gfx1250) — compile-verified
//
#include <hip/hip_runtime.h>
#include <hip/hip_bf16.h>

#define HIDDEN   128
#define HEADS    8
#define N_NODES  10000
#define N_COLS   40000
#define N_TOTAL  50000
#define N_EDGES  150000
#define N_SEG    (N_EDGES + N_TOTAL)   // edges + self loops
#define NEG_SLOPE 0.2f

typedef __attribute__((ext_vector_type(2))) float v2f;
typedef __attribute__((ext_vector_type(8))) float v8f;

// ---------------------------------------------------------------------------
// Specialized f32 WMMA GEMM for N=K=128: C[M,128] (+)= A[M,128] * B[128,128].
// Block = 256 threads = 8 waves; each wave computes a 16x128 output strip
// (8 accumulator tiles), so one A fragment feeds 8 v_wmma ops.
// B is staged in LDS in k-pair interleaved layout with padding so each
// B fragment is one conflict-free ds_load_b64:
//   lds[((k>>1)*BPAD + n)*2 + (k&1)]
// Pair-row stride BPAD=144 => upper half-wave (k+2) hits banks 32..63 while
// lower half-wave uses banks 0..31.  Staged in two K=64 chunks (36 KB LDS).
// ---------------------------------------------------------------------------
#define BPAD 144
#define KCHUNK 64

__global__ __launch_bounds__(256)
void wmma_gemm_nk128(const float* __restrict__ A,
                     const float* __restrict__ B, int ldb,
                     float* __restrict__ C,
                     int tilesM, int accumulate)
{
    __shared__ float bl[(KCHUNK / 2) * BPAD * 2];   // 36,864 B

    const int t    = (int)threadIdx.x;
    const int wave = t >> 5;
    const int lane = t & 31;
    const int tm   = (int)blockIdx.x * 8 + wave;
    const bool active = (tm < tilesM);              // wave-uniform
    const int hi = lane >> 4;                       // 0: lanes 0-15, 1: lanes 16-31
    const int lo = lane & 15;

    v8f acc[8];
    if (active) {
        if (accumulate) {
#pragma unroll
            for (int tn = 0; tn < 8; ++tn)
#pragma unroll
                for (int v = 0; v < 8; ++v)
                    acc[tn][v] = C[(size_t)(tm * 16 + v + 8 * hi) * 128 + tn * 16 + lo];
        } else {
#pragma unroll
            for (int tn = 0; tn < 8; ++tn)
#pragma unroll
                for (int v = 0; v < 8; ++v) acc[tn][v] = 0.0f;
        }
    }

    const float* arowp = A + (size_t)(tm * 16 + lo) * 128;

    for (int kc = 0; kc < 128; kc += KCHUNK) {
        // ---- stage B[kc .. kc+64) into LDS, coalesced global reads ----
#pragma unroll
        for (int i = 0; i < (KCHUNK * 128) / 256; ++i) {   // 32 iters
            int flat = i * 256 + t;                        // 0..8191
            int k = flat >> 7;                             // 0..63
            int n = flat & 127;
            bl[((k >> 1) * BPAD + n) * 2 + (k & 1)] = B[(size_t)(kc + k) * ldb + n];
        }
        __syncthreads();

        if (active) {
            for (int k0 = 0; k0 < KCHUNK; k0 += 4) {
                const int ka = k0 + 2 * hi;                // lane-group K split
                v2f a;
                a.x = arowp[kc + ka];
                a.y = arowp[kc + ka + 1];
                const int kp = ka >> 1;
#pragma unroll
                for (int tn = 0; tn < 8; ++tn) {
                    v2f b = *(const v2f*)&bl[(kp * BPAD + tn * 16 + lo) * 2];
                    acc[tn] = __builtin_amdgcn_wmma_f32_16x16x4_f32(
                                  false, a, false, b, (short)0, acc[tn], false, false);
                }
            }
        }
        __syncthreads();
    }

    if (active) {
#pragma unroll
        for (int tn = 0; tn < 8; ++tn)
#pragma unroll
            for (int v = 0; v < 8; ++v)
                C[(size_t)(tm * 16 + v + 8 * hi) * 128 + tn * 16 + lo] = acc[tn][v];
    }
}

// ---------------------------------------------------------------------------
// Input projection: x = relu(node_f @ Wn + bn) ++ relu(col_f @ Wc + bc)
// ---------------------------------------------------------------------------
__global__ void proj_kernel(const float* __restrict__ nf, const float* __restrict__ cf,
                            const float* __restrict__ Wn, const float* __restrict__ bn,
                            const float* __restrict__ Wc, const float* __restrict__ bc,
                            float* __restrict__ x)
{
    int idx = blockIdx.x * blockDim.x + threadIdx.x;
    if (idx >= N_TOTAL * HIDDEN) return;
    int row = idx >> 7;
    int j   = idx & 127;
    float s;
    if (row < N_NODES) {
        s = bn[j];
        const float* r = nf + (size_t)row * 16;
#pragma unroll
        for (int k = 0; k < 16; ++k) s += r[k] * Wn[k * HIDDEN + j];
    } else {
        s = bc[j];
        const float* r = cf + (size_t)(row - N_NODES) * 32;
#pragma unroll
        for (int k = 0; k < 32; ++k) s += r[k] * Wc[k * HIDDEN + j];
    }
    x[idx] = s > 0.0f ? s : 0.0f;
}

// ---------------------------------------------------------------------------
// a_src[n] = dot(h[n,:], att_src); a_dst[n] = dot(h[n,:], att_dst)
// One wave32 per node, 4 dims per lane, shuffle reduction.
// ---------------------------------------------------------------------------
__global__ void attn_dots(const float* __restrict__ h,
                          const float* __restrict__ att_src,
                          const float* __restrict__ att_dst,
                          float* __restrict__ a_src, float* __restrict__ a_dst, int n)
{
    int wave = (int)((blockIdx.x * blockDim.x + threadIdx.x) >> 5);
    int lane = (int)(threadIdx.x & 31);
    if (wave >= n) return;
    const float* hr = h + (size_t)wave * HIDDEN;
    float s0 = 0.0f, s1 = 0.0f;
#pragma unroll
    for (int i = 0; i < 4; ++i) {
        int d = lane + 32 * i;
        float hv = hr[d];
        s0 += hv * att_src[d];
        s1 += hv * att_dst[d];
    }
#pragma unroll
    for (int off = 16; off > 0; off >>= 1) {
        s0 += __shfl_down(s0, off, 32);
        s1 += __shfl_down(s1, off, 32);
    }
    if (lane == 0) { a_src[wave] = s0; a_dst[wave] = s1; }
}

// ---------------------------------------------------------------------------
// Edge helpers: segment e < N_EDGES is a real edge, else self-loop (e-N_EDGES).
// ---------------------------------------------------------------------------
__device__ __forceinline__ void edge_sd(const int* __restrict__ edges, int e,
                                        int& s, int& d)
{
    if (e < N_EDGES) { s = edges[e]; d = edges[N_EDGES + e]; }
    else             { s = d = e - N_EDGES; }
}

__device__ __forceinline__ unsigned flt_enc(float f)
{
    unsigned u = __float_as_uint(f);
    return u ^ ((u >> 31) ? 0xFFFFFFFFu : 0x80000000u);   // monotonic encoding
}
__device__ __forceinline__ float flt_dec(unsigned e)
{
    unsigned u = e ^ ((e & 0x80000000u) ? 0x80000000u : 0xFFFFFFFFu);
    return __uint_as_float(u);
}

__global__ void edge_max(const int* __restrict__ edges,
                         const float* __restrict__ a_src,
                         const float* __restrict__ a_dst,
                         unsigned* __restrict__ amax)
{
    int e = blockIdx.x * blockDim.x + threadIdx.x;
    if (e >= N_SEG) return;
    int s, d; edge_sd(edges, e, s, d);
    float al = a_src[s] + a_dst[d];
    al = al > 0.0f ? al : al * NEG_SLOPE;
    atomicMax(&amax[d], flt_enc(al));
}

__global__ void edge_exp(const int* __restrict__ edges,
                         const float* __restrict__ a_src,
                         const float* __restrict__ a_dst,
                         const unsigned* __restrict__ amax,
                         float* __restrict__ denom, float* __restrict__ ex)
{
    int e = blockIdx.x * blockDim.x + threadIdx.x;
    if (e >= N_SEG) return;
    int s, d; edge_sd(edges, e, s, d);
    float al = a_src[s] + a_dst[d];
    al = al > 0.0f ? al : al * NEG_SLOPE;
    float v = expf(al - flt_dec(amax[d]));
    ex[e] = v;
    atomicAdd(&denom[d], v);
}

// agg[dst,:] += h[src,:] * (ex[e] / (denom[dst]+1e-16)); one wave per edge.
__global__ void edge_agg(const int* __restrict__ edges,
                         const float* __restrict__ ex,
                         const float* __restrict__ denom,
                         const float* __restrict__ h,
                         float* __restrict__ agg)
{
    int wave = (int)((blockIdx.x * blockDim.x + threadIdx.x) >> 5);
    int lane = (int)(threadIdx.x & 31);
    if (wave >= N_SEG) return;
    int s, d; edge_sd(edges, wave, s, d);
    float w = ex[wave] / (denom[d] + 1e-16f);
    const float* hs = h + (size_t)s * HIDDEN;
    float* ad = agg + (size_t)d * HIDDEN;
#pragma unroll
    for (int i = 0; i < 4; ++i) {
        int dim = lane + 32 * i;
        atomicAdd(&ad[dim], hs[dim] * w);
    }
}

__global__ void bias_relu(float* __restrict__ y, const float* __restrict__ b,
                          int total, int cols)
{
    int idx = blockIdx.x * blockDim.x + threadIdx.x;
    if (idx >= total) return;
    float v = y[idx] + b[idx % cols];
    y[idx] = v > 0.0f ? v : 0.0f;
}

__global__ void fill_f32(float* __restrict__ p, float v, int n)
{
    int idx = blockIdx.x * blockDim.x + threadIdx.x;
    if (idx < n) p[idx] = v;
}

// ---------------------------------------------------------------------------
// Output MLP: out[row] = relu(x3[row] @ Wo1 + bo1) @ Wo2 + bo2 (rows 0..9999)
// ---------------------------------------------------------------------------
__global__ void out_mlp(const float* __restrict__ x3,
                        const float* __restrict__ Wo1, const float* __restrict__ bo1,
                        const float* __restrict__ Wo2, const float* __restrict__ bo2,
                        float* __restrict__ out)
{
    __shared__ float sh[64];
    int row = blockIdx.x;
    int j   = threadIdx.x;
    const float* xr = x3 + (size_t)row * HIDDEN;
    float s = bo1[j];
    for (int k = 0; k < HIDDEN; ++k) s += xr[k] * Wo1[k * 64 + j];
    s = s > 0.0f ? s : 0.0f;
    sh[j] = s * Wo2[j];
    __syncthreads();
    for (int off = 32; off > 0; off >>= 1) {
        if (j < off) sh[j] += sh[j + off];
        __syncthreads();
    }
    if (j == 0) out[row] = sh[0] + bo2[0];
}

// ---------------------------------------------------------------------------
// Host orchestration
// ---------------------------------------------------------------------------
extern "C" void kernel_launch(void* const* d_in, const int* in_sizes, int n_in,
                              void* d_out, int out_size, void* d_ws, size_t ws_size,
                              hipStream_t stream)
{
    const float* node_f   = (const float*)d_in[0];
    const float* col_f    = (const float*)d_in[1];
    const int*   edges    = (const int*)  d_in[2];
    const float* Wn       = (const float*)d_in[4];
    const float* bn       = (const float*)d_in[5];
    const float* Wc       = (const float*)d_in[6];
    const float* bc       = (const float*)d_in[7];
    const float* W1       = (const float*)d_in[8];   // [128, 1024]
    const float* att_s1   = (const float*)d_in[9];   // [8, 128]
    const float* att_d1   = (const float*)d_in[10];
    const float* b1       = (const float*)d_in[11];  // [1024]
    const float* W2       = (const float*)d_in[12];  // [1024, 128]
    const float* att_s2   = (const float*)d_in[13];  // [1, 128]
    const float* att_d2   = (const float*)d_in[14];
    const float* b2       = (const float*)d_in[15];  // [128]
    const float* Wo1      = (const float*)d_in[16];  // [128, 64]
    const float* bo1      = (const float*)d_in[17];
    const float* Wo2      = (const float*)d_in[18];  // [64, 1]
    const float* bo2      = (const float*)d_in[19];
    float* out            = (float*)d_out;

    // Workspace layout (floats): 4 * 25.6 MB feature buffers + small arrays
    const size_t NF = (size_t)N_TOTAL * HIDDEN;      // 6.4M floats
    float*    x     = (float*)d_ws;                  // [50000,128] projected features
    float*    hbuf  = x     + NF;                    // per-head h, later conv2 agg
    float*    agg   = hbuf  + NF;                    // per-head aggregation / x2_head
    float*    h2    = agg   + NF;                    // conv2 pre-attention features
    float*    a_src = h2    + NF;                    // [50000]
    float*    a_dst = a_src + N_TOTAL;
    unsigned* amax  = (unsigned*)(a_dst + N_TOTAL);  // [50000] encoded max
    float*    denom = (float*)(amax + N_TOTAL);      // [50000]
    float*    ex    = denom + N_TOTAL;               // [200000]

    const int B256 = 256;
    const int tilesM = N_TOTAL / 16;                              // 3125
    const int gNF    = (int)((NF + B256 - 1) / B256);             // 25000
    const int gSeg   = (N_SEG + B256 - 1) / B256;
    const int gSegW  = (N_SEG * 32 + B256 - 1) / B256;
    const int gNode  = (N_TOTAL * 32 + B256 - 1) / B256;
    const int gGemm  = (tilesM + 7) / 8;                          // 391 blocks
    const int gInit  = (2 * N_TOTAL + B256 - 1) / B256;

    // 1) input projections -> x
    proj_kernel<<<gNF, B256, 0, stream>>>(node_f, col_f, Wn, bn, Wc, bc, x);

    // 2) h2 accumulator = 0
    fill_f32<<<gNF, B256, 0, stream>>>(h2, 0.0f, (int)NF);

    // 3) GATConv1 per head, fused with GEMM2 partial accumulation
    for (int head = 0; head < HEADS; ++head) {
        // h_head = x @ W1[:, head*128 : head*128+128]
        wmma_gemm_nk128<<<gGemm, B256, 0, stream>>>(
            x, W1 + head * HIDDEN, HEADS * HIDDEN, hbuf, tilesM, 0);

        attn_dots<<<gNode, B256, 0, stream>>>(
            hbuf, att_s1 + head * HIDDEN, att_d1 + head * HIDDEN, a_src, a_dst, N_TOTAL);

        // amax (enc 0) and denom (0.0f) are bitwise-zero inits, contiguous
        fill_f32<<<gInit, B256, 0, stream>>>((float*)amax, 0.0f, 2 * N_TOTAL);

        edge_max<<<gSeg, B256, 0, stream>>>(edges, a_src, a_dst, amax);
        edge_exp<<<gSeg, B256, 0, stream>>>(edges, a_src, a_dst, amax, denom, ex);

        fill_f32<<<gNF, B256, 0, stream>>>(agg, 0.0f, (int)NF);
        edge_agg<<<gSegW, B256, 0, stream>>>(edges, ex, denom, hbuf, agg);

        // x2_head = relu(agg + b1_head)
        bias_relu<<<gNF, B256, 0, stream>>>(agg, b1 + head * HIDDEN, (int)NF, HIDDEN);

        // h2 += x2_head @ W2[head*128 : head*128+128, :]
        wmma_gemm_nk128<<<gGemm, B256, 0, stream>>>(
            agg, W2 + (size_t)head * HIDDEN * HIDDEN, HIDDEN, h2, tilesM, 1);
    }

    // 4) GATConv2 (heads=1). flip(edges) only reverses order: segment ops invariant.
    attn_dots<<<gNode, B256, 0, stream>>>(h2, att_s2, att_d2, a_src, a_dst, N_TOTAL);
    fill_f32<<<gInit, B256, 0, stream>>>((float*)amax, 0.0f, 2 * N_TOTAL);
    edge_max<<<gSeg, B256, 0, stream>>>(edges, a_src, a_dst, amax);
    edge_exp<<<gSeg, B256, 0, stream>>>(edges, a_src, a_dst, amax, denom, ex);
    fill_f32<<<gNF, B256, 0, stream>>>(hbuf, 0.0f, (int)NF);      // reuse hbuf as agg2
    edge_agg<<<gSegW, B256, 0, stream>>>(edges, ex, denom, h2, hbuf);
    bias_relu<<<gNF, B256, 0, stream>>>(hbuf, b2, (int)NF, HIDDEN); // x3 = relu(agg2+b2)

    // 5) output MLP on node rows only
    out_mlp<<<N_NODES, 64, 0, stream>>>(hbuf, Wo1, bo1, Wo2, bo2, out);
}